// BidirMambaEncoder_8409545966074
// MI455X (gfx1250) — compile-verified
//
#include <hip/hip_runtime.h>
#include <math.h>

// ---------------- model constants ----------------
#define D_MODEL  512
#define FFN_DIM  896
#define D_STATE  16
#define D_CONV   4
#define D_INNER  1024
#define DT_RANK  32
#define BATCH    2
#define TLEN     512
#define NTOK     (BATCH * TLEN)      // 1024 tokens

typedef __attribute__((ext_vector_type(2))) float v2f;
typedef __attribute__((ext_vector_type(8))) float v8f;

#define WMMA_F32(a, b, c) \
    __builtin_amdgcn_wmma_f32_16x16x4_f32(false, (a), false, (b), (short)0, (c), false, false)

// ---------------- device math helpers ----------------
__device__ __forceinline__ float siluf(float x) { return x / (1.0f + expf(-x)); }
__device__ __forceinline__ float geluf(float x) {
    return 0.5f * x * (1.0f + erff(x * 0.7071067811865476f));
}
__device__ __forceinline__ float softplusf(float x) {
    return (x > 20.0f) ? x : log1pf(expf(x));
}

// ---------------- fp32 WMMA GEMM:  C = act(A @ W^T + bias) (+ residual) (* mask) ------------
// A: M x K (row-major, lda), W: N x K (row-major, ldw), C: M x N (row-major, ldc)
// One wave computes a 32x64 tile: 8 accumulators; per K-step of 4: 6 v2f loads feed 8 WMMAs.
// Next K-step operands are prefetched into registers before the current WMMA group so the
// loads overlap matrix-op issue (software pipelining, depth 1).
// act: 0 = none, 1 = gelu, 2 = softplus. lengths != null => zero masked positions (rev for bwd).
__global__ void k_gemm_wmma(const float* __restrict__ A, int lda,
                            const float* __restrict__ W, int ldw,
                            float* __restrict__ C, int ldc,
                            const float* __restrict__ bias,
                            const float* __restrict__ res, int ldr,
                            int M, int N, int K, int act,
                            const int* __restrict__ lengths, int rev)
{
    const int wavesPerBlock = blockDim.x >> 5;
    const int wave   = blockIdx.x * wavesPerBlock + (threadIdx.x >> 5);
    const int tilesN = N >> 6;               // 64-wide wave tiles
    const int tilesM = M >> 5;               // 32-row wave tiles
    if (wave >= tilesM * tilesN) return;     // wave-uniform exit (EXEC stays all-ones)

    const int tm = wave / tilesN;
    const int tn = wave % tilesN;
    const int lane = threadIdx.x & 31;
    const int g = lane >> 4;                 // lane group (0: lanes 0-15, 1: lanes 16-31)
    const int h = lane & 15;                 // row/col within group
    const int m0 = tm << 5;
    const int n0 = tn << 6;

    // A f32 16x4 layout: lane g*16+m, vgpr v -> A[m][2g+v]   (float2 per lane per step)
    const float* Ap0 = A + (size_t)(m0 + h) * lda + 2 * g;
    const float* Ap1 = Ap0 + (size_t)16 * lda;
    // B = W^T, f32 4x16 layout: lane g*16+n, vgpr v -> W[n][2g+v]
    const float* Wp  = W + (size_t)(n0 + h) * ldw + 2 * g;
    const size_t ws16 = (size_t)16 * ldw;

    v8f acc00 = {}, acc01 = {}, acc02 = {}, acc03 = {};
    v8f acc10 = {}, acc11 = {}, acc12 = {}, acc13 = {};

    // prologue loads (k = 0)
    v2f a0 = *(const v2f*)(Ap0);
    v2f a1 = *(const v2f*)(Ap1);
    v2f b0 = *(const v2f*)(Wp);
    v2f b1 = *(const v2f*)(Wp + ws16);
    v2f b2 = *(const v2f*)(Wp + 2 * ws16);
    v2f b3 = *(const v2f*)(Wp + 3 * ws16);

    for (int k = 4; k < K; k += 4) {
        // prefetch next K-step before issuing current WMMAs
        v2f na0 = *(const v2f*)(Ap0 + k);
        v2f na1 = *(const v2f*)(Ap1 + k);
        v2f nb0 = *(const v2f*)(Wp + k);
        v2f nb1 = *(const v2f*)(Wp + ws16 + k);
        v2f nb2 = *(const v2f*)(Wp + 2 * ws16 + k);
        v2f nb3 = *(const v2f*)(Wp + 3 * ws16 + k);

        acc00 = WMMA_F32(a0, b0, acc00);
        acc01 = WMMA_F32(a0, b1, acc01);
        acc02 = WMMA_F32(a0, b2, acc02);
        acc03 = WMMA_F32(a0, b3, acc03);
        acc10 = WMMA_F32(a1, b0, acc10);
        acc11 = WMMA_F32(a1, b1, acc11);
        acc12 = WMMA_F32(a1, b2, acc12);
        acc13 = WMMA_F32(a1, b3, acc13);

        a0 = na0; a1 = na1; b0 = nb0; b1 = nb1; b2 = nb2; b3 = nb3;
    }
    // tail K-step
    acc00 = WMMA_F32(a0, b0, acc00);
    acc01 = WMMA_F32(a0, b1, acc01);
    acc02 = WMMA_F32(a0, b2, acc02);
    acc03 = WMMA_F32(a0, b3, acc03);
    acc10 = WMMA_F32(a1, b0, acc10);
    acc11 = WMMA_F32(a1, b1, acc11);
    acc12 = WMMA_F32(a1, b2, acc12);
    acc13 = WMMA_F32(a1, b3, acc13);

    // C/D 16x16 layout: vgpr r, lane g*16+n -> D[8g + r][n]
    v8f accs[2][4] = {{acc00, acc01, acc02, acc03}, {acc10, acc11, acc12, acc13}};
    #pragma unroll
    for (int half = 0; half < 2; ++half) {
        #pragma unroll
        for (int j = 0; j < 4; ++j) {
            const int n = n0 + j * 16 + h;
            const float bz = bias ? bias[n] : 0.0f;
            #pragma unroll
            for (int r = 0; r < 8; ++r) {
                const int m = m0 + 16 * half + 8 * g + r;
                float v = accs[half][j][r] + bz;
                if (act == 1)      v = geluf(v);
                else if (act == 2) v = softplusf(v);
                if (res) v += res[(size_t)m * ldr + n];
                if (lengths) {
                    const int b = m >> 9;                   // TLEN = 512
                    const int t = m & (TLEN - 1);
                    const int pos = rev ? (TLEN - 1 - t) : t;
                    if (pos >= lengths[b]) v = 0.0f;
                }
                C[(size_t)m * ldc + n] = v;
            }
        }
    }
}

// ---------------- LayerNorm: one block (256 thr) per token row of 512 ----------------
__global__ void k_ln(const float* __restrict__ in, const float* __restrict__ w,
                     const float* __restrict__ b, float* __restrict__ out)
{
    __shared__ float red[256];
    const int row = blockIdx.x;
    const int tid = threadIdx.x;
    const float* xr = in + (size_t)row * D_MODEL;
    float v0 = xr[tid], v1 = xr[tid + 256];

    red[tid] = v0 + v1;
    __syncthreads();
    for (int s = 128; s > 0; s >>= 1) { if (tid < s) red[tid] += red[tid + s]; __syncthreads(); }
    const float mu = red[0] * (1.0f / 512.0f);
    __syncthreads();

    const float d0 = v0 - mu, d1 = v1 - mu;
    red[tid] = d0 * d0 + d1 * d1;
    __syncthreads();
    for (int s = 128; s > 0; s >>= 1) { if (tid < s) red[tid] += red[tid + s]; __syncthreads(); }
    const float rstd = rsqrtf(red[0] * (1.0f / 512.0f) + 1e-5f);

    float* orow = out + (size_t)row * D_MODEL;
    orow[tid]       = d0 * rstd * w[tid]       + b[tid];
    orow[tid + 256] = d1 * rstd * w[tid + 256] + b[tid + 256];
}

// ---------------- causal conv1d (K=4) + SiLU; input = xm half of xz ----------------
__global__ void k_conv_silu(const float* __restrict__ xz,
                            const float* __restrict__ cw,
                            const float* __restrict__ cb,
                            float* __restrict__ out)
{
    const int idx = blockIdx.x * blockDim.x + threadIdx.x;
    if (idx >= NTOK * D_INNER) return;
    const int e   = idx & (D_INNER - 1);
    const int tok = idx >> 10;               // b*TLEN + t
    const int t   = tok & (TLEN - 1);
    float s = cb[e];
    #pragma unroll
    for (int k = 0; k < D_CONV; ++k) {
        const int tt = t - (D_CONV - 1) + k;
        if (tt >= 0)
            s += xz[(size_t)(tok - (D_CONV - 1) + k) * (2 * D_INNER) + e] * cw[e * D_CONV + k];
    }
    out[idx] = siluf(s);
}

// ---------------- selective scan + SiLU gate: one block per batch, one thread per channel ----
// u = conv output, delta = softplus(dt_proj), dbl holds B_ (cols 32..47) and C_ (cols 48..63),
// xz supplies the z gate (second half). Writes yg = y * silu(z) directly.
__global__ void __launch_bounds__(1024) k_scan(const float* __restrict__ u,
                                               const float* __restrict__ delta,
                                               const float* __restrict__ dbl,
                                               const float* __restrict__ A_log,
                                               const float* __restrict__ Dv,
                                               const float* __restrict__ xz,
                                               float* __restrict__ yg)
{
    const int b = blockIdx.x;
    const int e = threadIdx.x;               // 0..1023
    float Arow[D_STATE];
    #pragma unroll
    for (int n = 0; n < D_STATE; ++n) Arow[n] = -expf(A_log[e * D_STATE + n]);
    const float Dd = Dv[e];
    float h[D_STATE];
    #pragma unroll
    for (int n = 0; n < D_STATE; ++n) h[n] = 0.0f;

    __shared__ float sBC[32];
    for (int t = 0; t < TLEN; ++t) {
        const int row = b * TLEN + t;
        if (threadIdx.x < 32) sBC[threadIdx.x] = dbl[(size_t)row * 64 + 32 + threadIdx.x];
        __syncthreads();
        const float dlt = delta[(size_t)row * D_INNER + e];
        const float uu  = u[(size_t)row * D_INNER + e];
        const float du  = dlt * uu;
        float acc = 0.0f;
        #pragma unroll
        for (int n = 0; n < D_STATE; ++n) {
            h[n] = expf(dlt * Arow[n]) * h[n] + du * sBC[n];
            acc += h[n] * sBC[16 + n];
        }
        const float z = xz[(size_t)row * (2 * D_INNER) + D_INNER + e];
        yg[(size_t)row * D_INNER + e] = (acc + uu * Dd) * siluf(z);
        __syncthreads();
    }
}

// ---------------- init: x + sinusoidal PE, produce fwd and time-reversed bwd streams --------
__global__ void k_init(const float* __restrict__ x, float* __restrict__ xf, float* __restrict__ xb)
{
    const int idx = blockIdx.x * blockDim.x + threadIdx.x;
    if (idx >= NTOK * D_MODEL) return;
    const int d   = idx & (D_MODEL - 1);
    const int tok = idx >> 9;
    const int b = tok >> 9;
    const int t = tok & (TLEN - 1);
    const float freq = expf(-(float)(d & ~1) * (9.210340371976184f / (float)D_MODEL));
    const float ang  = (float)t * freq;
    const float pe   = (d & 1) ? cosf(ang) : sinf(ang);
    const float v = x[idx] + pe;
    xf[idx] = v;
    xb[((size_t)(b * TLEN + (TLEN - 1 - t))) * D_MODEL + d] = v;
}

// ---------------- final: out = x_fwd + reverse_t(x_bwd) ----------------
__global__ void k_final(const float* __restrict__ xf, const float* __restrict__ xb,
                        float* __restrict__ out)
{
    const int idx = blockIdx.x * blockDim.x + threadIdx.x;
    if (idx >= NTOK * D_MODEL) return;
    const int d   = idx & (D_MODEL - 1);
    const int tok = idx >> 9;
    const int b = tok >> 9;
    const int t = tok & (TLEN - 1);
    out[idx] = xf[idx] + xb[((size_t)(b * TLEN + (TLEN - 1 - t))) * D_MODEL + d];
}

// ---------------- host side ----------------
struct LayerP {
    const float *ln1_w, *ln1_b, *ln2_w, *ln2_b;
    const float *in_proj_w, *conv_w, *conv_b, *x_proj_w;
    const float *dt_proj_w, *dt_proj_b, *A_log, *D, *out_proj_w;
    const float *ffn_w1, *ffn_b1, *ffn_w2, *ffn_b2;
    const float *ln0_w, *ln0_b;
};

static void launch_gemm(hipStream_t s, const float* A, int lda, const float* W, int ldw,
                        float* C, int ldc, const float* bias, const float* res, int ldr,
                        int M, int N, int K, int act,
                        const int* lengths = nullptr, int rev = 0)
{
    const int waves = (M / 32) * (N / 64);
    const int wpb = 4;
    dim3 grid((waves + wpb - 1) / wpb), block(32 * wpb);
    hipLaunchKernelGGL(k_gemm_wmma, grid, block, 0, s,
                       A, lda, W, ldw, C, ldc, bias, res, ldr, M, N, K, act, lengths, rev);
}

extern "C" void kernel_launch(void* const* d_in, const int* in_sizes, int n_in,
                              void* d_out, int out_size, void* d_ws, size_t ws_size,
                              hipStream_t stream)
{
    (void)in_sizes; (void)n_in; (void)out_size; (void)ws_size;

    const float* x_in    = (const float*)d_in[0];
    const int*   lengths = (const int*)d_in[1];

    // params flattened in dict-insertion order: fwd[0], fwd[1], bwd[0], bwd[1];
    // per layer: ln1_w, ln1_b, ln2_w, ln2_b, in_proj_w, conv_w, conv_b, x_proj_w,
    //            dt_proj_w, dt_proj_b, A_log, D, out_proj_w, ffn_w1, ffn_b1,
    //            ffn_w2, ffn_b2, [ln0_w, ln0_b (layer 0 only, appended last)]
    LayerP P[2][2];
    int c = 2;
    for (int dir = 0; dir < 2; ++dir) {
        for (int layer = 0; layer < 2; ++layer) {
            LayerP& p = P[dir][layer];
            p.ln1_w      = (const float*)d_in[c++];
            p.ln1_b      = (const float*)d_in[c++];
            p.ln2_w      = (const float*)d_in[c++];
            p.ln2_b      = (const float*)d_in[c++];
            p.in_proj_w  = (const float*)d_in[c++];
            p.conv_w     = (const float*)d_in[c++];
            p.conv_b     = (const float*)d_in[c++];
            p.x_proj_w   = (const float*)d_in[c++];
            p.dt_proj_w  = (const float*)d_in[c++];
            p.dt_proj_b  = (const float*)d_in[c++];
            p.A_log      = (const float*)d_in[c++];
            p.D          = (const float*)d_in[c++];
            p.out_proj_w = (const float*)d_in[c++];
            p.ffn_w1     = (const float*)d_in[c++];
            p.ffn_b1     = (const float*)d_in[c++];
            p.ffn_w2     = (const float*)d_in[c++];
            p.ffn_b2     = (const float*)d_in[c++];
            if (layer == 0) {
                p.ln0_w = (const float*)d_in[c++];
                p.ln0_b = (const float*)d_in[c++];
            } else {
                p.ln0_w = nullptr; p.ln0_b = nullptr;
            }
        }
    }

    // workspace carve-out (floats)
    float* ws = (float*)d_ws;
    size_t off = 0;
    auto alloc = [&](size_t n) { float* p = ws + off; off += n; return p; };
    float* xdir[2] = { alloc((size_t)NTOK * D_MODEL), alloc((size_t)NTOK * D_MODEL) };
    float* xn    = alloc((size_t)NTOK * D_MODEL);
    float* xz    = alloc((size_t)NTOK * 2 * D_INNER);
    float* xmc   = alloc((size_t)NTOK * D_INNER);
    float* dbl   = alloc((size_t)NTOK * 64);
    float* delta = alloc((size_t)NTOK * D_INNER);
    float* yg    = alloc((size_t)NTOK * D_INNER);
    float* hffn  = alloc((size_t)NTOK * FFN_DIM);

    const int EB = 256;
    hipLaunchKernelGGL(k_init, dim3((NTOK * D_MODEL) / EB), dim3(EB), 0, stream,
                       x_in, xdir[0], xdir[1]);

    for (int dir = 0; dir < 2; ++dir) {
        float* x = xdir[dir];
        for (int layer = 0; layer < 2; ++layer) {
            const LayerP& p = P[dir][layer];
            if (layer == 0)
                hipLaunchKernelGGL(k_ln, dim3(NTOK), dim3(256), 0, stream, x, p.ln0_w, p.ln0_b, x);

            // mamba block
            hipLaunchKernelGGL(k_ln, dim3(NTOK), dim3(256), 0, stream, x, p.ln1_w, p.ln1_b, xn);
            launch_gemm(stream, xn, D_MODEL, p.in_proj_w, D_MODEL, xz, 2 * D_INNER,
                        nullptr, nullptr, 0, NTOK, 2 * D_INNER, D_MODEL, 0);
            hipLaunchKernelGGL(k_conv_silu, dim3((NTOK * D_INNER) / EB), dim3(EB), 0, stream,
                               xz, p.conv_w, p.conv_b, xmc);
            launch_gemm(stream, xmc, D_INNER, p.x_proj_w, D_INNER, dbl, 64,
                        nullptr, nullptr, 0, NTOK, 64, D_INNER, 0);
            launch_gemm(stream, dbl, 64, p.dt_proj_w, DT_RANK, delta, D_INNER,
                        p.dt_proj_b, nullptr, 0, NTOK, D_INNER, DT_RANK, 2 /*softplus*/);
            hipLaunchKernelGGL(k_scan, dim3(BATCH), dim3(D_INNER), 0, stream,
                               xmc, delta, dbl, p.A_log, p.D, xz, yg);
            launch_gemm(stream, yg, D_INNER, p.out_proj_w, D_INNER, x, D_MODEL,
                        nullptr, x, D_MODEL, NTOK, D_MODEL, D_INNER, 0);   // x += yg @ W^T

            // FFN block (mask fused into second GEMM epilogue)
            hipLaunchKernelGGL(k_ln, dim3(NTOK), dim3(256), 0, stream, x, p.ln2_w, p.ln2_b, xn);
            launch_gemm(stream, xn, D_MODEL, p.ffn_w1, D_MODEL, hffn, FFN_DIM,
                        p.ffn_b1, nullptr, 0, NTOK, FFN_DIM, D_MODEL, 1 /*gelu*/);
            launch_gemm(stream, hffn, FFN_DIM, p.ffn_w2, FFN_DIM, x, D_MODEL,
                        p.ffn_b2, x, D_MODEL, NTOK, D_MODEL, FFN_DIM, 0,
                        lengths, dir /* rev for bwd stream */);            // x = mask*(x + h@W^T + b)

        }
    }

    hipLaunchKernelGGL(k_final, dim3((NTOK * D_MODEL) / EB), dim3(EB), 0, stream,
                       xdir[0], xdir[1], (float*)d_out);
}